// Loss_54322746359839
// MI455X (gfx1250) — compile-verified
//
#include <hip/hip_runtime.h>
#include <math.h>

// Negative log-likelihood: loss = -sum_i log(pred[i, tgt[i]])
// B = 262144 rows, V = 1024 cols (taken from in_sizes at launch for generality).
//
// Gather-bound kernel: ~32 MiB effective HBM traffic (one cacheline per row).
// Two-pass deterministic reduction; wave-level sums use V_WMMA_F32_16X16X4_F32
// (ones-matrix reduction trick) to exercise the CDNA5 matrix path.

#define K1_BLOCKS  1024
#define K1_THREADS 256
#define K2_THREADS 256

typedef __attribute__((ext_vector_type(2))) float v2f;
typedef __attribute__((ext_vector_type(8))) float v8f;

// Sum a float across all 32 lanes of a wave using one WMMA.
// A (16x4):  lane L<16 -> A[L][0]=s, A[L][1]=0 ; lane L>=16 -> A[L-16][2]=s, A[L-16][3]=0
// B (4x16):  all ones  =>  D[m][n] = s_m + s_{m+16}
// D (16x16): VGPR j = rows {j, j+8}; per-lane sum of the 8 accum regs gives the
// half-wave sum; fold halves with one shfl_xor(16). EXEC must be all-1s here.
__device__ __forceinline__ float wave32_sum_wmma(float s) {
    v2f a;    a[0] = s;    a[1] = 0.0f;
    v2f ones; ones[0] = 1.0f; ones[1] = 1.0f;
    v8f c = {};
    c = __builtin_amdgcn_wmma_f32_16x16x4_f32(
        /*neg_a=*/false, a, /*neg_b=*/false, ones,
        /*c_mod=*/(short)0, c, /*reuse_a=*/false, /*reuse_b=*/false);
    float t = ((c[0] + c[1]) + (c[2] + c[3])) + ((c[4] + c[5]) + (c[6] + c[7]));
    t += __shfl_xor(t, 16, 32);   // combine the two half-wave row groups
    return t;                     // every lane now holds the full wave sum
}

__global__ __launch_bounds__(K1_THREADS)
void nll_partial_kernel(const float* __restrict__ pred,
                        const int*   __restrict__ tgt,
                        float*       __restrict__ part,
                        int n_rows, int n_cols) {
    const int tid    = blockIdx.x * K1_THREADS + threadIdx.x;
    const int stride = K1_BLOCKS * K1_THREADS;

    float s = 0.0f;
    for (int i = tid; i < n_rows; i += stride) {
        // Targets are streamed once; gathered prediction lines are used once:
        // nontemporal so 32 MiB of gather traffic doesn't churn L2.
        int   t = __builtin_nontemporal_load(tgt + i);
        float p = __builtin_nontemporal_load(pred + (size_t)i * (size_t)n_cols + (size_t)t);
        s += logf(p);
    }

    // Threads reconverge here (EXEC all-1s) -> safe for WMMA.
    float wsum = wave32_sum_wmma(s);

    __shared__ float smem[K1_THREADS / 32];
    const int lane = threadIdx.x & 31;
    const int wid  = threadIdx.x >> 5;
    if (lane == 0) smem[wid] = wsum;
    __syncthreads();

    if (wid == 0) {
        float v = (lane < (K1_THREADS / 32)) ? smem[lane] : 0.0f;
        v += __shfl_xor(v, 4, 32);
        v += __shfl_xor(v, 2, 32);
        v += __shfl_xor(v, 1, 32);
        if (lane == 0) part[blockIdx.x] = v;
    }
}

__global__ __launch_bounds__(K2_THREADS)
void nll_final_kernel(const float* __restrict__ part,
                      float*       __restrict__ out,
                      int n_parts) {
    float s = 0.0f;
    for (int i = threadIdx.x; i < n_parts; i += K2_THREADS)
        s += part[i];

    float wsum = wave32_sum_wmma(s);

    __shared__ float smem[K2_THREADS / 32];
    const int lane = threadIdx.x & 31;
    const int wid  = threadIdx.x >> 5;
    if (lane == 0) smem[wid] = wsum;
    __syncthreads();

    if (wid == 0) {
        float v = (lane < (K2_THREADS / 32)) ? smem[lane] : 0.0f;
        v += __shfl_xor(v, 4, 32);
        v += __shfl_xor(v, 2, 32);
        v += __shfl_xor(v, 1, 32);
        if (lane == 0) out[0] = -v;   // overwrite (poison-safe, deterministic)
    }
}

extern "C" void kernel_launch(void* const* d_in, const int* in_sizes, int n_in,
                              void* d_out, int out_size, void* d_ws, size_t ws_size,
                              hipStream_t stream) {
    const float* pred = (const float*)d_in[0];   // [B, V] float32
    const int*   tgt  = (const int*)d_in[1];     // [B] int
    float*       out  = (float*)d_out;           // scalar
    float*       part = (float*)d_ws;            // K1_BLOCKS partial sums (4 KB)

    const int n_rows = in_sizes[1];              // B
    const int n_cols = in_sizes[0] / in_sizes[1];// V

    nll_partial_kernel<<<K1_BLOCKS, K1_THREADS, 0, stream>>>(pred, tgt, part, n_rows, n_cols);
    nll_final_kernel<<<1, K2_THREADS, 0, stream>>>(part, out, K1_BLOCKS);
}